// QuantumLayer_57123065037199
// MI455X (gfx1250) — compile-verified
//
#include <hip/hip_runtime.h>

typedef float v2f __attribute__((ext_vector_type(2)));
typedef float v8f __attribute__((ext_vector_type(8)));

__device__ __forceinline__ v8f wmma_f32(v2f a, v2f b, v8f c) {
    return __builtin_amdgcn_wmma_f32_16x16x4_f32(false, a, false, b, (short)0, c, false, false);
}

// ---------------------------------------------------------------------------
// Kernel 1: build per-layer Kronecker factors A (wires 0-4) and B (wires 5-9).
// ws layout per layer l (4096 floats): Ar[1024] Ai[1024] Br[1024] Bi[1024]
// ---------------------------------------------------------------------------
__global__ __launch_bounds__(256) void build_unitaries(const float* __restrict__ w1,
                                                       const float* __restrict__ w2,
                                                       float* __restrict__ U) {
    // Rot(phi,theta,omega) = [[ep*c, -conj(em)*s],[em*s, conj(ep)*c]]
    // ep = exp(-0.5i(phi+omega)), em = exp(-0.5i(phi-omega))
    __shared__ float mre[4][10][2][2];
    __shared__ float mim[4][10][2][2];

    const int tid = threadIdx.x;
    if (tid < 40) {
        const int l = tid / 10;
        const int j = tid % 10;
        const float* w = (l < 3) ? (w1 + (l * 10 + j) * 3) : (w2 + j * 3);
        const float phi = w[0], theta = w[1], omega = w[2];
        const float c = cosf(0.5f * theta);
        const float s = sinf(0.5f * theta);
        float sa, ca, sb, cb;
        sincosf(-0.5f * (phi + omega), &sa, &ca);   // ep = (ca, sa)
        sincosf(-0.5f * (phi - omega), &sb, &cb);   // em = (cb, sb)
        mre[l][j][0][0] =  ca * c;  mim[l][j][0][0] =  sa * c;   // ep*c
        mre[l][j][0][1] = -cb * s;  mim[l][j][0][1] =  sb * s;   // -conj(em)*s
        mre[l][j][1][0] =  cb * s;  mim[l][j][1][0] =  sb * s;   // em*s
        mre[l][j][1][1] =  ca * c;  mim[l][j][1][1] = -sa * c;   // conj(ep)*c
    }
    __syncthreads();

    // 4 layers x 2 matrices x 1024 entries = 8192 complex entries
    for (int it = 0; it < 32; ++it) {
        const int idx = tid + it * 256;
        const int l   = idx >> 11;
        const int rem = idx & 2047;
        const int mat = rem >> 10;          // 0 = A (wires 0-4), 1 = B (wires 5-9)
        const int e   = rem & 1023;
        const int r   = e >> 5;
        const int cc  = e & 31;
        float pr = 1.f, pi = 0.f;
        const int j0 = mat ? 5 : 0;
        #pragma unroll
        for (int jj = 0; jj < 5; ++jj) {
            const int rb = (r  >> (4 - jj)) & 1;
            const int cb = (cc >> (4 - jj)) & 1;
            const float ar = mre[l][j0 + jj][rb][cb];
            const float ai = mim[l][j0 + jj][rb][cb];
            const float nr = pr * ar - pi * ai;
            const float ni = pr * ai + pi * ar;
            pr = nr; pi = ni;
        }
        float* base = U + l * 4096 + mat * 2048;
        base[e]        = pr;
        base[e + 1024] = pi;
    }
}

// ---------------------------------------------------------------------------
// Kernel 2: one block (4 waves) per batch element.
// State as 32x32 complex matrix in LDS; each layer: S' = A * S * B^T via
// v_wmma_f32_16x16x4_f32 tiles, then CZ-ring sign (+ reupload phase).
// ---------------------------------------------------------------------------
__global__ __launch_bounds__(128) void qsim(const float* __restrict__ x,
                                            const float* __restrict__ U,
                                            float* __restrict__ out) {
    __shared__ float Sr[1024], Si[1024];
    __shared__ float Cr[1024], Ci[1024];
    __shared__ float Ar[1024], Ai[1024], Br[1024], Bi[1024];
    __shared__ float prr[32], pri[32], pcr[32], pci[32];
    __shared__ float red[128][10];

    const int tid  = threadIdx.x;
    const int b    = blockIdx.x;
    const int lane = tid & 31;
    const int wid  = tid >> 5;
    const int ti   = wid >> 1;            // tile row (0..1)
    const int tj   = wid & 1;             // tile col (0..1)

    // |0...0> : S[0][0] = 1
    for (int i = tid; i < 1024; i += 128) { Sr[i] = (i == 0) ? 1.f : 0.f; Si[i] = 0.f; }

    float xb[10];
    #pragma unroll
    for (int j = 0; j < 10; ++j) xb[j] = x[b * 10 + j];

    // Data-reuploading phase diag factorizes: exp(-0.5i x.signs) = pr[row]*pc[col]
    if (tid < 64) {
        const int  idx = tid & 31;
        const bool isC = tid >= 32;
        float a = 0.f;
        #pragma unroll
        for (int jj = 0; jj < 5; ++jj) {
            const int bit = (idx >> (4 - jj)) & 1;
            a += xb[isC ? 5 + jj : jj] * (1.f - 2.f * (float)bit);
        }
        a *= -0.5f;
        float sv, cv; sincosf(a, &sv, &cv);
        if (isC) { pcr[idx] = cv; pci[idx] = sv; }
        else     { prr[idx] = cv; pri[idx] = sv; }
    }
    __syncthreads();

    const int mrow = ti * 16 + (lane & 15);      // A-fragment row
    const int ncol = tj * 16 + (lane & 15);      // B-fragment col
    const int ksub = (lane >> 4) << 1;           // f32 WMMA K sublane offset

    for (int l = 0; l < 4; ++l) {
        // stage layer factors into LDS (L2-broadcast across 2048 blocks)
        const float* base = U + l * 4096;
        for (int i = tid; i < 1024; i += 128) {
            Ar[i] = base[i];        Ai[i] = base[1024 + i];
            Br[i] = base[2048 + i]; Bi[i] = base[3072 + i];
        }
        __syncthreads();

        // -------- C = A * S (complex, 16x16 tile per wave) --------
        {
            v8f cr = {}; v8f ci = {};
            #pragma unroll
            for (int kk = 0; kk < 8; ++kk) {
                const int k = kk * 4 + ksub;
                v2f aR, aI, aIn, bR, bI;
                aR.x = Ar[mrow * 32 + k];  aR.y = Ar[mrow * 32 + k + 1];
                aI.x = Ai[mrow * 32 + k];  aI.y = Ai[mrow * 32 + k + 1];
                aIn.x = -aI.x;             aIn.y = -aI.y;
                bR.x = Sr[k * 32 + ncol];  bR.y = Sr[(k + 1) * 32 + ncol];
                bI.x = Si[k * 32 + ncol];  bI.y = Si[(k + 1) * 32 + ncol];
                cr = wmma_f32(aR,  bR, cr);
                cr = wmma_f32(aIn, bI, cr);
                ci = wmma_f32(aR,  bI, ci);
                ci = wmma_f32(aI,  bR, ci);
            }
            #pragma unroll
            for (int v = 0; v < 8; ++v) {
                const int row = ti * 16 + v + ((lane >> 4) << 3);
                const int col = tj * 16 + (lane & 15);
                Cr[row * 32 + col] = cr[v];
                Ci[row * 32 + col] = ci[v];
            }
        }
        __syncthreads();

        // -------- T = C * B^T, then diagonal (CZ ring, + phase for l<3) ------
        {
            v8f tr = {}; v8f tv = {};
            #pragma unroll
            for (int kk = 0; kk < 8; ++kk) {
                const int k = kk * 4 + ksub;
                v2f aR, aI, aIn, bR, bI;
                aR.x = Cr[mrow * 32 + k];  aR.y = Cr[mrow * 32 + k + 1];
                aI.x = Ci[mrow * 32 + k];  aI.y = Ci[mrow * 32 + k + 1];
                aIn.x = -aI.x;             aIn.y = -aI.y;
                bR.x = Br[ncol * 32 + k];  bR.y = Br[ncol * 32 + k + 1]; // B^T[k][n]
                bI.x = Bi[ncol * 32 + k];  bI.y = Bi[ncol * 32 + k + 1];
                tr = wmma_f32(aR,  bR, tr);
                tr = wmma_f32(aIn, bI, tr);
                tv = wmma_f32(aR,  bI, tv);
                tv = wmma_f32(aI,  bR, tv);
            }
            #pragma unroll
            for (int v = 0; v < 8; ++v) {
                const int row = ti * 16 + v + ((lane >> 4) << 3);
                const int col = tj * 16 + (lane & 15);
                const int k   = row * 32 + col;
                const unsigned u = (unsigned)k;
                // CZ ring parity: adjacent-wire pairs + (wire9,wire0) closure
                const int par = __popc(u & (u >> 1)) + (int)((u & 1u) & (u >> 9));
                const float sgn = (par & 1) ? -1.f : 1.f;
                float pRe = sgn, pIm = 0.f;
                if (l < 3) {
                    const float a1 = prr[row], b1 = pri[row];
                    const float a2 = pcr[col], b2 = pci[col];
                    pRe = sgn * (a1 * a2 - b1 * b2);
                    pIm = sgn * (a1 * b2 + b1 * a2);
                }
                const float er = tr[v], ei = tv[v];
                Sr[k] = er * pRe - ei * pIm;
                Si[k] = er * pIm + ei * pRe;
            }
        }
        __syncthreads();
    }

    // -------- probs -> <Z_j>, deterministic two-stage reduction --------
    float acc[10];
    #pragma unroll
    for (int j = 0; j < 10; ++j) acc[j] = 0.f;
    for (int i = tid; i < 1024; i += 128) {
        const float p = Sr[i] * Sr[i] + Si[i] * Si[i];
        #pragma unroll
        for (int j = 0; j < 10; ++j)
            acc[j] += ((i >> (9 - j)) & 1) ? -p : p;
    }
    #pragma unroll
    for (int j = 0; j < 10; ++j) red[tid][j] = acc[j];
    __syncthreads();
    if (tid < 10) {
        float s = 0.f;
        for (int t = 0; t < 128; ++t) s += red[t][tid];
        out[b * 10 + tid] = s;
    }
}

// ---------------------------------------------------------------------------
extern "C" void kernel_launch(void* const* d_in, const int* in_sizes, int n_in,
                              void* d_out, int out_size, void* d_ws, size_t ws_size,
                              hipStream_t stream) {
    const float* x  = (const float*)d_in[0];   // [B, 10]
    const float* w1 = (const float*)d_in[1];   // [3, 1, 10, 3]
    const float* w2 = (const float*)d_in[2];   // [1, 10, 3]
    float* out = (float*)d_out;                // [B, 10]
    float* U   = (float*)d_ws;                 // 4 * 4096 floats = 64 KB

    const int B = in_sizes[0] / 10;

    hipLaunchKernelGGL(build_unitaries, dim3(1), dim3(256), 0, stream, w1, w2, U);
    hipLaunchKernelGGL(qsim, dim3(B), dim3(128), 0, stream, x, U, out);
}